// GAT_10041633538818
// MI455X (gfx1250) — compile-verified
//
#include <hip/hip_runtime.h>
#include <hip/hip_bf16.h>
#include <math.h>
#include <stdint.h>

#define HEADS 4

typedef __attribute__((ext_vector_type(2))) float v2f;
typedef __attribute__((ext_vector_type(8))) float v8f;

// -------------------------------------------------------------------------
// GEMM: Hout[N,Fout] = X[N,128] @ W[128,Fout]  (fp32, V_WMMA_F32_16X16X4_F32)
// One block = 16 output rows; X strip staged in LDS via CDNA5 async
// global->LDS copy (ASYNCcnt); one wave per 16x16 output tile.
// -------------------------------------------------------------------------
__global__ void gat_gemm_wmma(const float* __restrict__ X,
                              const float* __restrict__ W,
                              float* __restrict__ Hout,
                              int Nn, int Fout) {
    __shared__ float As[16 * 128];
    const int tid  = threadIdx.x;
    const int row0 = blockIdx.x * 16;
    const bool fullStrip = (row0 + 16 <= Nn);   // uniform per block

    if (fullStrip) {
        // CDNA5 async copy: 16 rows x 512 B = 8 KB, B128 per lane, 2 issues
        // per thread (256 threads). Generic LDS pointer low 32 bits = LDS
        // offset (aperture lives in the high half), which is what the
        // async-op VDST operand wants.
        const uint32_t ldsBase = (uint32_t)(uintptr_t)(&As[0]);
        const uint64_t gBase   = (uint64_t)(uintptr_t)(X + (size_t)row0 * 128);
#pragma unroll
        for (int it = 0; it < 2; ++it) {
            const uint32_t i = (uint32_t)tid + (uint32_t)it * 256u;
            uint32_t laddr = ldsBase + i * 16u;
            uint64_t gaddr = gBase + (uint64_t)i * 16u;
            asm volatile("global_load_async_to_lds_b128 %0, %1, off"
                         :: "v"(laddr), "v"(gaddr) : "memory");
        }
        asm volatile("s_wait_asynccnt 0" ::: "memory");
    } else {
        for (int i = tid; i < 16 * 128; i += blockDim.x) {
            int r = row0 + (i >> 7);
            As[i] = (r < Nn) ? X[(size_t)r * 128 + (i & 127)] : 0.0f;
        }
    }
    __syncthreads();

    const int wave   = tid >> 5;
    const int lane   = tid & 31;
    const int m      = lane & 15;           // M (A) / N (B) index
    const int koff   = (lane >> 4) << 1;    // lanes 0-15: K=0,1 ; lanes 16-31: K=2,3
    const int nWaves = blockDim.x >> 5;
    const int nTiles = Fout >> 4;

    for (int ct = wave; ct < nTiles; ct += nWaves) {
        const int col0 = ct << 4;
        v8f acc = {};
#pragma unroll 4
        for (int k = 0; k < 128; k += 4) {
            // A fragment (16x4): lane holds A[m][k+koff], A[m][k+koff+1]
            v2f a = *(const v2f*)(&As[m * 128 + k + koff]);
            // B fragment (4x16): lane holds W[k+koff][col0+m], W[k+koff+1][col0+m]
            v2f b;
            b.x = W[(size_t)(k + koff)     * Fout + col0 + m];
            b.y = W[(size_t)(k + koff + 1) * Fout + col0 + m];
            acc = __builtin_amdgcn_wmma_f32_16x16x4_f32(
                false, a, false, b, (short)0, acc, false, false);
        }
        // D layout: VGPR v -> row (lane<16 ? v : v+8), col = col0 + (lane&15)
        const int rbase = row0 + ((lane >> 4) << 3);
        const int dcol  = col0 + m;
        if (fullStrip) {
            float* outp = Hout + (size_t)rbase * Fout + dcol;
#pragma unroll
            for (int v = 0; v < 8; ++v) outp[(size_t)v * Fout] = acc[v];
        } else {
#pragma unroll
            for (int v = 0; v < 8; ++v) {
                int r = rbase + v;
                if (r < Nn) Hout[(size_t)r * Fout + dcol] = acc[v];
            }
        }
    }
}

// -------------------------------------------------------------------------
// Per-node attention logits: alpha_s[n,h] = <h[n,h,:], a_src[h,:]> etc.
// -------------------------------------------------------------------------
__global__ void gat_node_alpha(const float* __restrict__ Hbuf,
                               const float* __restrict__ a_src,
                               const float* __restrict__ a_dst,
                               float* __restrict__ alpha_s,
                               float* __restrict__ alpha_d,
                               int Nn, int C) {
    int t = blockIdx.x * blockDim.x + threadIdx.x;
    if (t >= Nn * HEADS) return;
    int n  = t >> 2;
    int hh = t & 3;
    const float* hp  = Hbuf + (size_t)n * (HEADS * C) + hh * C;
    const float* av  = a_src + hh * C;
    const float* bv  = a_dst + hh * C;
    float ss = 0.0f, sd = 0.0f;
    for (int c = 0; c < C; ++c) {
        float v = hp[c];
        ss += v * av[c];
        sd += v * bv[c];
    }
    alpha_s[t] = ss;
    alpha_d[t] = sd;
}

__global__ void fill_f32(float* __restrict__ p, float v, size_t n) {
    size_t i      = (size_t)blockIdx.x * blockDim.x + threadIdx.x;
    size_t stride = (size_t)gridDim.x * blockDim.x;
    for (; i < n; i += stride) p[i] = v;
}

__device__ inline void atomicMaxF(float* addr, float val) {
    // works for buffers initialized to -inf
    if (val >= 0.0f)
        atomicMax((int*)addr, __float_as_int(val));
    else
        atomicMin((unsigned int*)addr, (unsigned int)__float_as_int(val));
}

// -------------------------------------------------------------------------
// Edge pass 1: segment max of leaky_relu(alpha_s[src]+alpha_d[dst]) by dst
// -------------------------------------------------------------------------
__global__ void gat_edge_max(const float* __restrict__ alpha_s,
                             const float* __restrict__ alpha_d,
                             const int* __restrict__ ei0,
                             const int* __restrict__ ei1,
                             float* __restrict__ mbuf, int EA, int E) {
    int e = blockIdx.x * blockDim.x + threadIdx.x;
    if (e >= EA) return;
    int s = (e < E) ? ei0[e] : (e - E);
    int d = (e < E) ? ei1[e] : (e - E);
    const float4 as = *(const float4*)(alpha_s + (size_t)s * 4);
    const float4 ad = *(const float4*)(alpha_d + (size_t)d * 4);
    float ev[4] = {as.x + ad.x, as.y + ad.y, as.z + ad.z, as.w + ad.w};
#pragma unroll
    for (int hh = 0; hh < 4; ++hh) {
        float v = ev[hh];
        v = (v > 0.0f) ? v : 0.2f * v;
        atomicMaxF(mbuf + (size_t)d * 4 + hh, v);
    }
}

// -------------------------------------------------------------------------
// Edge pass 2: ex = exp(e - m[dst]); denom[dst] += ex; store ex
// -------------------------------------------------------------------------
__global__ void gat_edge_exp(const float* __restrict__ alpha_s,
                             const float* __restrict__ alpha_d,
                             const int* __restrict__ ei0,
                             const int* __restrict__ ei1,
                             const float* __restrict__ mbuf,
                             float* __restrict__ denom,
                             float* __restrict__ exbuf, int EA, int E) {
    int e = blockIdx.x * blockDim.x + threadIdx.x;
    if (e >= EA) return;
    int s = (e < E) ? ei0[e] : (e - E);
    int d = (e < E) ? ei1[e] : (e - E);
    const float4 as = *(const float4*)(alpha_s + (size_t)s * 4);
    const float4 ad = *(const float4*)(alpha_d + (size_t)d * 4);
    const float4 mm = *(const float4*)(mbuf + (size_t)d * 4);
    float ev[4] = {as.x + ad.x, as.y + ad.y, as.z + ad.z, as.w + ad.w};
    float mv[4] = {mm.x, mm.y, mm.z, mm.w};
#pragma unroll
    for (int hh = 0; hh < 4; ++hh) {
        float v  = ev[hh];
        v        = (v > 0.0f) ? v : 0.2f * v;
        float ex = expf(v - mv[hh]);
        exbuf[(size_t)e * 4 + hh] = ex;
        atomicAdd(denom + (size_t)d * 4 + hh, ex);
    }
}

// -------------------------------------------------------------------------
// Edge pass 3: acc[dst] += h[src] * (ex/denom[dst]) ; one wave32 per edge
// -------------------------------------------------------------------------
__global__ void gat_edge_agg(const float* __restrict__ Hbuf,
                             const float* __restrict__ exbuf,
                             const float* __restrict__ denom,
                             const int* __restrict__ ei0,
                             const int* __restrict__ ei1,
                             float* __restrict__ acc, int EA, int E, int C) {
    const int lane = threadIdx.x & 31;
    const int e    = (int)(((size_t)blockIdx.x * blockDim.x + threadIdx.x) >> 5);
    if (e >= EA) return;
    int s = (e < E) ? ei0[e] : (e - E);
    int d = (e < E) ? ei1[e] : (e - E);
    float al[4];
#pragma unroll
    for (int hh = 0; hh < 4; ++hh)
        al[hh] = exbuf[(size_t)e * 4 + hh] / denom[(size_t)d * 4 + hh];
    const int HC = HEADS * C;
    for (int cbase = lane * 4; cbase < HC; cbase += 128) {
        const float4 hv = *(const float4*)(Hbuf + (size_t)s * HC + cbase);
        float vv[4] = {hv.x, hv.y, hv.z, hv.w};
#pragma unroll
        for (int j = 0; j < 4; ++j) {
            int c  = cbase + j;
            int hh = c / C;
            atomicAdd(acc + (size_t)d * HC + c, vv[j] * al[hh]);
        }
    }
}

// -------------------------------------------------------------------------
// Epilogues
// -------------------------------------------------------------------------
__global__ void gat_finalize12(const float* __restrict__ acc,
                               const float* __restrict__ bias,
                               const float* __restrict__ resid,  // may be null
                               float* __restrict__ xout, int total) {
    int i = blockIdx.x * blockDim.x + threadIdx.x;
    if (i >= total) return;
    int c   = i & 127;
    float v = acc[i] + bias[c];
    if (resid) v += resid[i];
    xout[i] = (v > 0.0f) ? v : (expf(v) - 1.0f);   // ELU
}

__global__ void gat_finalize3(const float* __restrict__ acc,
                              const float* __restrict__ bias,
                              float* __restrict__ out, int Nn, int NC) {
    int i = blockIdx.x * blockDim.x + threadIdx.x;
    if (i >= Nn * NC) return;
    int n = i / NC;
    int j = i - n * NC;
    float s = 0.0f;
#pragma unroll
    for (int hh = 0; hh < HEADS; ++hh)
        s += acc[(size_t)n * (HEADS * NC) + hh * NC + j];
    out[i] = s * (1.0f / HEADS) + bias[j];
}

// -------------------------------------------------------------------------
extern "C" void kernel_launch(void* const* d_in, const int* in_sizes, int n_in,
                              void* d_out, int out_size, void* d_ws, size_t ws_size,
                              hipStream_t stream) {
    const float* x       = (const float*)d_in[0];
    const int*   eidx    = (const int*)d_in[1];
    const float* W1      = (const float*)d_in[2];
    const float* a_src1  = (const float*)d_in[3];
    const float* a_dst1  = (const float*)d_in[4];
    const float* b1      = (const float*)d_in[5];
    const float* W2      = (const float*)d_in[6];
    const float* a_src2  = (const float*)d_in[7];
    const float* a_dst2  = (const float*)d_in[8];
    const float* b2      = (const float*)d_in[9];
    const float* W3      = (const float*)d_in[10];
    const float* a_src3  = (const float*)d_in[11];
    const float* a_dst3  = (const float*)d_in[12];
    const float* b3      = (const float*)d_in[13];

    const int N  = in_sizes[0] / 128;      // 50000
    const int E  = in_sizes[1] / 2;        // 800000
    const int NC = in_sizes[13];           // 40
    const int EA = E + N;                  // edges incl. self-loops
    const int* ei0 = eidx;                 // src
    const int* ei1 = eidx + E;             // dst

    // workspace layout (floats)
    float* ws = (float*)d_ws;
    size_t off = 0;
    float* h_buf = ws + off; off += (size_t)N * 160;   // h = X@W (max Fout)
    float* accb  = ws + off; off += (size_t)N * 160;   // aggregation accumulator
    float* x1    = ws + off; off += (size_t)N * 128;
    float* x2    = ws + off; off += (size_t)N * 128;
    float* as_   = ws + off; off += (size_t)N * 4;
    float* ad_   = ws + off; off += (size_t)N * 4;
    float* mb    = ws + off; off += (size_t)N * 4;
    float* dn    = ws + off; off += (size_t)N * 4;
    float* exb   = ws + off; off += (size_t)EA * 4;

    const int gGemm = (N + 15) / 16;
    const int gNA   = (N * HEADS + 255) / 256;
    const int gEdge = (EA + 255) / 256;
    const int gAgg  = (EA + 7) / 8;        // 8 wave32 edges per 256-thread block

    auto run_layer = [&](const float* Xin, const float* W, const float* a_s,
                         const float* a_d, int C, int Fout) {
        gat_gemm_wmma<<<gGemm, 256, 0, stream>>>(Xin, W, h_buf, N, Fout);
        gat_node_alpha<<<gNA, 256, 0, stream>>>(h_buf, a_s, a_d, as_, ad_, N, C);
        fill_f32<<<256, 256, 0, stream>>>(mb, -INFINITY, (size_t)N * 4);
        fill_f32<<<256, 256, 0, stream>>>(dn, 0.0f, (size_t)N * 4);
        fill_f32<<<2048, 256, 0, stream>>>(accb, 0.0f, (size_t)N * Fout);
        gat_edge_max<<<gEdge, 256, 0, stream>>>(as_, ad_, ei0, ei1, mb, EA, E);
        gat_edge_exp<<<gEdge, 256, 0, stream>>>(as_, ad_, ei0, ei1, mb, dn, exb, EA, E);
        gat_edge_agg<<<gAgg, 256, 0, stream>>>(h_buf, exb, dn, ei0, ei1, accb, EA, E, C);
    };

    // Layer 1: x1 = elu(gat(x) + b1)
    run_layer(x, W1, a_src1, a_dst1, 32, 128);
    gat_finalize12<<<(N * 128 + 255) / 256, 256, 0, stream>>>(accb, b1, nullptr, x1, N * 128);

    // Layer 2: x2 = elu(gat(x1) + b2 + x1)
    run_layer(x1, W2, a_src2, a_dst2, 32, 128);
    gat_finalize12<<<(N * 128 + 255) / 256, 256, 0, stream>>>(accb, b2, x1, x2, N * 128);

    // Layer 3: out = mean_heads(gat(x2)) + b3
    run_layer(x2, W3, a_src3, a_dst3, NC, HEADS * NC);
    gat_finalize3<<<(N * NC + 255) / 256, 256, 0, stream>>>(accb, b3, (float*)d_out, N, NC);
}